// SparseVoxelEncoder_58050777973221
// MI455X (gfx1250) — compile-verified
//
#include <hip/hip_runtime.h>

#define DDIM 32
#define VOXEL_INV 4.0f          // 1 / 0.25
#define WAVES_PER_BLOCK 8
#define THREADS_PER_BLOCK 256   // 8 wave32 waves
#define PTS_PER_WAVE 8

#if __has_builtin(__builtin_amdgcn_global_load_async_to_lds_b128) && \
    __has_builtin(__builtin_amdgcn_s_wait_asynccnt)
#define USE_ASYNC_GATHER 1
#else
#define USE_ASYNC_GATHER 0
#endif

#define AS1 __attribute__((address_space(1)))
#define AS3 __attribute__((address_space(3)))

typedef int v4i __attribute__((ext_vector_type(4)));

// generic->AS1 (identity mapping for global memory), typed as v4i per the
// builtin's signature revealed by the round-2 diagnostic.
__device__ __forceinline__ AS1 v4i* gptr_v4(const void* p) {
    return (AS1 v4i*)(unsigned long long)p;
}
// generic->AS3 via u32 truncation: low 32 bits of a flat LDS address are the
// LDS offset, and AS3 pointers are 32-bit.
__device__ __forceinline__ AS3 v4i* lptr_v4(void* p) {
    return (AS3 v4i*)(unsigned int)(unsigned long long)p;
}

// corner c -> (qx,qy,qz) = ((c>>2)&1,(c>>1)&1,c&1); w = prod(q ? f : 1-f)
__device__ __forceinline__ void trilerp_weights(float fx, float fy, float fz,
                                                float w[8]) {
    const float gx = 1.0f - fx, gy = 1.0f - fy, gz = 1.0f - fz;
    const float a00 = gx * gy, a01 = gx * fy, a10 = fx * gy, a11 = fx * fy;
    w[0] = a00 * gz; w[1] = a00 * fz;
    w[2] = a01 * gz; w[3] = a01 * fz;
    w[4] = a10 * gz; w[5] = a10 * fz;
    w[6] = a11 * gz; w[7] = a11 * fz;
}

#if USE_ASYNC_GATHER
// Compute weights for point p and launch the async gather of its 8 corner
// rows (8 x 32 f32 = 1 KB) into `dst`. Two B128 issues: each issue moves
// 4 rows (lane octet o handles corner o, (lane&7)*16B within the row).
__device__ __forceinline__ void stage_point_async(
    int p, int lane,
    const float* __restrict__ sxyz, const float* __restrict__ pxyz,
    const float* __restrict__ vals, const int* __restrict__ sidx,
    const int* __restrict__ pfeat, float* dst, float w[8])
{
    const int vi = __builtin_amdgcn_readfirstlane(sidx[p]);
    const float fx = (sxyz[3 * p + 0] - pxyz[3 * vi + 0]) * VOXEL_INV + 0.5f;
    const float fy = (sxyz[3 * p + 1] - pxyz[3 * vi + 1]) * VOXEL_INV + 0.5f;
    const float fz = (sxyz[3 * p + 2] - pxyz[3 * vi + 2]) * VOXEL_INV + 0.5f;
    trilerp_weights(fx, fy, fz, w);

    const int c  = lane >> 3;            // corner group 0..3 for this lane
    const int fo = (lane & 7) * 4;       // float offset inside the row
    const int r0 = pfeat[8 * vi + c];        // corners 0..3
    const int r1 = pfeat[8 * vi + c + 4];    // corners 4..7
    const float* g0 = vals + (size_t)r0 * DDIM + fo;
    const float* g1 = vals + (size_t)r1 * DDIM + fo;
    float* d0 = dst + c * DDIM + fo;
    float* d1 = dst + (c + 4) * DDIM + fo;
    __builtin_amdgcn_global_load_async_to_lds_b128(gptr_v4(g0), lptr_v4(d0), 0, 0);
    __builtin_amdgcn_global_load_async_to_lds_b128(gptr_v4(g1), lptr_v4(d1), 0, 0);
}
#endif

__global__ __launch_bounds__(THREADS_PER_BLOCK)
void sparse_voxel_trilerp_kernel(const float* __restrict__ sxyz,   // [P,3]
                                 const float* __restrict__ pxyz,   // [H,3]
                                 const float* __restrict__ vals,   // [NEMB,32]
                                 const int*   __restrict__ sidx,   // [P]
                                 const int*   __restrict__ pfeat,  // [H,8]
                                 float*       __restrict__ out,    // [P,32]
                                 int P)
{
    const int lane  = threadIdx.x & 31;
    const int wv    = threadIdx.x >> 5;
    const int waveG = (int)blockIdx.x * WAVES_PER_BLOCK + wv;
    const int pbase = __builtin_amdgcn_readfirstlane(waveG * PTS_PER_WAVE);
    if (pbase >= P) return;
    const int nPts = (P - pbase < PTS_PER_WAVE) ? (P - pbase) : PTS_PER_WAVE;

#if USE_ASYNC_GATHER
    // 2 KB double buffer per wave -> 16 KB per block (of 320 KB/WGP).
    __shared__ float stage[WAVES_PER_BLOCK][2][8 * DDIM];
    float* const b0 = &stage[wv][0][0];
    float* const b1 = &stage[wv][1][0];

    float w_cur[8];
    stage_point_async(pbase, lane, sxyz, pxyz, vals, sidx, pfeat, b0, w_cur);

    for (int j = 0; j < nPts; ++j) {
        const bool have_nxt = (j + 1 < nPts);       // wave-uniform
        float w_nxt[8];
        if (have_nxt)
            stage_point_async(pbase + j + 1, lane, sxyz, pxyz, vals, sidx, pfeat,
                              ((j + 1) & 1) ? b1 : b0, w_nxt);

        // Retire batch j; batch j+1 (2 issues) may remain in flight.
        if (have_nxt) __builtin_amdgcn_s_wait_asynccnt(2);
        else          __builtin_amdgcn_s_wait_asynccnt(0);

        const float* b = (j & 1) ? b1 : b0;
        float acc =       w_cur[0] * b[0 * DDIM + lane];
        acc = fmaf(w_cur[1], b[1 * DDIM + lane], acc);
        acc = fmaf(w_cur[2], b[2 * DDIM + lane], acc);
        acc = fmaf(w_cur[3], b[3 * DDIM + lane], acc);
        acc = fmaf(w_cur[4], b[4 * DDIM + lane], acc);
        acc = fmaf(w_cur[5], b[5 * DDIM + lane], acc);
        acc = fmaf(w_cur[6], b[6 * DDIM + lane], acc);
        acc = fmaf(w_cur[7], b[7 * DDIM + lane], acc);
        out[(size_t)(pbase + j) * DDIM + lane] = acc;   // coalesced 128B store

        if (have_nxt) {
#pragma unroll
            for (int c = 0; c < 8; ++c) w_cur[c] = w_nxt[c];
        }
    }
#else
    // -------- fallback: direct VGPR gather (round-1 proven codegen) --------
    for (int j = 0; j < nPts; ++j) {
        const int p  = pbase + j;
        const int vi = __builtin_amdgcn_readfirstlane(sidx[p]);
        const float fx = (sxyz[3 * p + 0] - pxyz[3 * vi + 0]) * VOXEL_INV + 0.5f;
        const float fy = (sxyz[3 * p + 1] - pxyz[3 * vi + 1]) * VOXEL_INV + 0.5f;
        const float fz = (sxyz[3 * p + 2] - pxyz[3 * vi + 2]) * VOXEL_INV + 0.5f;
        float w[8];
        trilerp_weights(fx, fy, fz, w);

        const int* __restrict__ pf = pfeat + 8 * (size_t)vi;
        float v[8];
#pragma unroll
        for (int c = 0; c < 8; ++c) {
            const int row = __builtin_amdgcn_readfirstlane(pf[c]);
            v[c] = vals[(size_t)row * DDIM + lane];
        }
        float acc =       w[0] * v[0];
        acc = fmaf(w[1], v[1], acc);
        acc = fmaf(w[2], v[2], acc);
        acc = fmaf(w[3], v[3], acc);
        acc = fmaf(w[4], v[4], acc);
        acc = fmaf(w[5], v[5], acc);
        acc = fmaf(w[6], v[6], acc);
        acc = fmaf(w[7], v[7], acc);
        out[(size_t)p * DDIM + lane] = acc;
    }
#endif
}

extern "C" void kernel_launch(void* const* d_in, const int* in_sizes, int n_in,
                              void* d_out, int out_size, void* d_ws, size_t ws_size,
                              hipStream_t stream) {
    // setup_inputs() order:
    //   0: sampled_xyz [P,3] f32
    //   1: point_xyz   [H,3] f32
    //   2: values      [NEMB,32] f32
    //   3: sampled_idx [P]   int
    //   4: point_feats [H,8] int
    const float* sampled_xyz = (const float*)d_in[0];
    const float* point_xyz   = (const float*)d_in[1];
    const float* values      = (const float*)d_in[2];
    const int*   sampled_idx = (const int*)d_in[3];
    const int*   point_feats = (const int*)d_in[4];
    float* out = (float*)d_out;

    const int P = in_sizes[3];  // number of sampled points
    const int pts_per_block = WAVES_PER_BLOCK * PTS_PER_WAVE;  // 64
    const int blocks = (P + pts_per_block - 1) / pts_per_block;

    sparse_voxel_trilerp_kernel<<<blocks, THREADS_PER_BLOCK, 0, stream>>>(
        sampled_xyz, point_xyz, values, sampled_idx, point_feats, out, P);
}